// TripletLoss_55344948576712
// MI455X (gfx1250) — compile-verified
//
#include <hip/hip_runtime.h>
#include <hip/hip_bf16.h>

#define BN 384
#define DD 256
#define MARGIN 0.2f
#define NEG_INF_F (-1e30f)

typedef float v2f __attribute__((ext_vector_type(2)));
typedef float v8f __attribute__((ext_vector_type(8)));

// ---------------------------------------------------------------------------
// Exact JAX threefry2x32 -> uniform -> gumbel for element e of a flat array of
// n = BN^3 elements, key = jax.random.key(42) = {0, 42}.
// JAX pairs counters (b, b+n/2); out[e] = x0' for e < n/2 else x1' of block e-n/2.
// ---------------------------------------------------------------------------
__device__ __forceinline__ float gumbel_at(unsigned e, unsigned half) {
  unsigned b, pick;
  if (e < half) { b = e;        pick = 0u; }
  else          { b = e - half; pick = 1u; }
  unsigned x0 = b, x1 = b + half;
  const unsigned ks0 = 0u;
  const unsigned ks1 = 42u;
  const unsigned ks2 = 0u ^ 42u ^ 0x1BD11BDAu;
  x0 += ks0; x1 += ks1;
#define TF_RND(R) { x0 += x1; x1 = (x1 << (R)) | (x1 >> (32 - (R))); x1 ^= x0; }
  TF_RND(13) TF_RND(15) TF_RND(26) TF_RND(6)   x0 += ks1; x1 += ks2 + 1u;
  TF_RND(17) TF_RND(29) TF_RND(16) TF_RND(24)  x0 += ks2; x1 += ks0 + 2u;
  TF_RND(13) TF_RND(15) TF_RND(26) TF_RND(6)   x0 += ks0; x1 += ks1 + 3u;
  TF_RND(17) TF_RND(29) TF_RND(16) TF_RND(24)  x0 += ks1; x1 += ks2 + 4u;
  TF_RND(13) TF_RND(15) TF_RND(26) TF_RND(6)   x0 += ks2; x1 += ks0 + 5u;
#undef TF_RND
  unsigned bits = pick ? x1 : x0;
  // JAX uniform(minval=tiny, maxval=1): bits>>9 | 0x3f800000 viewed as float - 1
  float f = __uint_as_float((bits >> 9) | 0x3f800000u) - 1.0f;
  const float TINY = 1.17549435e-38f;
  float u = fmaxf(f * (1.0f - TINY) + TINY, TINY);
  return -logf(-logf(u));
}

// ---------------------------------------------------------------------------
// Kernel 0: zero the two accumulators (sum, count).
// ---------------------------------------------------------------------------
__global__ void tl_zero_kernel(float* sumAcc, unsigned* cntAcc) {
  *sumAcc = 0.0f;
  *cntAcc = 0u;
}

// ---------------------------------------------------------------------------
// Kernel 1: row squared norms. One wave per row.
// ---------------------------------------------------------------------------
__global__ __launch_bounds__(32)
void tl_norms_kernel(const float* __restrict__ F, float* __restrict__ norms) {
  const int row  = blockIdx.x;
  const int lane = threadIdx.x;
  float s = 0.0f;
  #pragma unroll
  for (int d = lane; d < DD; d += 32) {
    float x = F[row * DD + d];
    s += x * x;
  }
  #pragma unroll
  for (int off = 16; off > 0; off >>= 1) s += __shfl_xor(s, off, 32);
  if (lane == 0) norms[row] = s;
}

// ---------------------------------------------------------------------------
// Kernel 2: dist[i][j] via f32 WMMA Gram tiles.
// One wave computes one 16x16 tile of G = F F^T with 64 x V_WMMA_F32_16X16X4_F32,
// then dist = sqrt(max(ni + nj - 2G, 1e-11)).
// A 16x4 f32 layout: lanes 0-15 rows, VGPR0/1 = K {0,1}; lanes 16-31 = K {2,3}.
// B = F[tj]^T follows the identical per-lane float2 pattern with tj's rows.
// ---------------------------------------------------------------------------
__global__ __launch_bounds__(32)
void tl_dist_wmma_kernel(const float* __restrict__ F,
                         const float* __restrict__ norms,
                         float* __restrict__ dist) {
  const int ti   = blockIdx.y;
  const int tj   = blockIdx.x;
  const int lane = threadIdx.x;          // 0..31
  const int r    = lane & 15;
  const int kh   = (lane >> 4) * 2;      // 0 or 2
  const float* __restrict__ pa = F + (ti * 16 + r) * DD + kh;
  const float* __restrict__ pb = F + (tj * 16 + r) * DD + kh;

  v8f c = {};
  #pragma unroll
  for (int k = 0; k < DD; k += 4) {
    v2f a = *(const v2f*)(pa + k);
    v2f b = *(const v2f*)(pb + k);
    // (neg_a, A, neg_b, B, c_mod, C, reuse_a, reuse_b)
    c = __builtin_amdgcn_wmma_f32_16x16x4_f32(false, a, false, b,
                                              (short)0, c, false, false);
  }

  // C/D layout: lanes 0-15 -> M = v, N = lane; lanes 16-31 -> M = 8+v, N = lane-16
  const int n     = r;
  const int mbase = (lane >> 4) * 8;
  const int gj    = tj * 16 + n;
  const float nj  = norms[gj];
  #pragma unroll
  for (int v = 0; v < 8; ++v) {
    const int gi = ti * 16 + mbase + v;
    float s = norms[gi] + nj - 2.0f * c[v];
    s = fmaxf(s, 1e-11f);
    dist[gi * BN + gj] = sqrtf(s);
  }
}

// ---------------------------------------------------------------------------
// Kernel 3: semi-hard mining, pruned to pos_mask pairs (same label, p > i).
// One wave per (i,p); lanes stride the 384 negatives, exact gumbel on the fly,
// wave32 argmax reduction with first-index tie-break (matches jnp.argmax).
// ---------------------------------------------------------------------------
__global__ __launch_bounds__(32)
void tl_mine_kernel(const int* __restrict__ labels,
                    const int* __restrict__ epochp,
                    const float* __restrict__ dist,
                    float* __restrict__ sumAcc,
                    unsigned* __restrict__ cntAcc) {
  const int i = blockIdx.y;
  const int p = blockIdx.x;
  if (p <= i) return;
  const int li = labels[i];
  if (labels[p] != li) return;                 // pos_mask pruning

  const int  semiMode = (*epochp) > 3;
  const int  lane     = threadIdx.x;
  const float dpos    = dist[i * BN + p];
  const unsigned half  = (unsigned)BN * (unsigned)BN * (unsigned)BN / 2u;
  const unsigned ebase = ((unsigned)i * BN + (unsigned)p) * (unsigned)BN;

  float best  = -3.0e38f;
  int   bestk = BN;
  int   found = 0;

  #pragma unroll
  for (int k = lane; k < BN; k += 32) {
    const float dik = dist[i * BN + k];
    const int   neg = (labels[k] != li);
    int   sel;
    float logit;
    if (semiMode) {
      sel   = neg && (dik > dpos) && (dik < dpos + MARGIN);
      logit = sel ? -logf(dik) : NEG_INF_F;
    } else {
      sel   = neg;
      logit = sel ? 0.0f : NEG_INF_F;
    }
    const float score = logit + gumbel_at(ebase + (unsigned)k, half);
    if (score > best) { best = score; bestk = k; }
    found |= sel;
  }

  #pragma unroll
  for (int off = 16; off > 0; off >>= 1) {
    float ob = __shfl_xor(best,  off, 32);
    int   ok = __shfl_xor(bestk, off, 32);
    int   of = __shfl_xor(found, off, 32);
    found |= of;
    if (ob > best || (ob == best && ok < bestk)) { best = ob; bestk = ok; }
  }

  if (lane == 0 && found) {
    const float nd = dist[i * BN + bestk];
    const float pp = fmaxf(dpos - nd + MARGIN, 0.0f);
    atomicAdd(sumAcc, pp);
    atomicAdd(cntAcc, 1u);
  }
}

// ---------------------------------------------------------------------------
// Kernel 4: finalize mean.
// ---------------------------------------------------------------------------
__global__ void tl_finalize_kernel(const float* sumAcc, const unsigned* cntAcc,
                                   float* out) {
  const unsigned c = *cntAcc;
  out[0] = (c > 0u) ? (*sumAcc / (float)c) : 0.0f;
}

// ---------------------------------------------------------------------------
// Launch: zero -> norms -> WMMA dist -> mine -> finalize. Workspace layout:
// [0, 384*384) dist | [384*384, +384) norms | then sum (f32), count (u32).
// ---------------------------------------------------------------------------
extern "C" void kernel_launch(void* const* d_in, const int* in_sizes, int n_in,
                              void* d_out, int out_size, void* d_ws, size_t ws_size,
                              hipStream_t stream) {
  const float* features = (const float*)d_in[0];
  const int*   labels   = (const int*)d_in[1];
  const int*   epochp   = (const int*)d_in[2];
  float*       out      = (float*)d_out;

  float*    ws     = (float*)d_ws;
  float*    dist   = ws;                      // 384*384 floats
  float*    norms  = ws + BN * BN;            // 384 floats
  float*    sumAcc = ws + BN * BN + BN;       // 1 float
  unsigned* cntAcc = (unsigned*)(ws + BN * BN + BN + 1);

  tl_zero_kernel<<<1, 1, 0, stream>>>(sumAcc, cntAcc);
  tl_norms_kernel<<<BN, 32, 0, stream>>>(features, norms);
  tl_dist_wmma_kernel<<<dim3(BN / 16, BN / 16), 32, 0, stream>>>(features, norms, dist);
  tl_mine_kernel<<<dim3(BN, BN), 32, 0, stream>>>(labels, epochp, dist, sumAcc, cntAcc);
  tl_finalize_kernel<<<1, 1, 0, stream>>>(sumAcc, cntAcc, out);
}